// QuantizedParameter_34754875359404
// MI455X (gfx1250) — compile-verified
//
#include <hip/hip_runtime.h>
#include <stdint.h>

// Q4_0 dequant: 18-byte blocks (fp16 scale + 16 packed nibbles) -> 32 fp32.
#define QK             32
#define BLOCK_BYTES    18
#define THREADS        256
#define BLOCKS_PER_WG  256
#define CHUNK_BYTES    (BLOCKS_PER_WG * BLOCK_BYTES)   // 4608
#define CHUNK_DWORDS   (CHUNK_BYTES / 4)               // 1152

typedef float    f32x4 __attribute__((ext_vector_type(4)));
typedef uint32_t u32x4 __attribute__((ext_vector_type(4)));
typedef int      v4i   __attribute__((ext_vector_type(4)));
typedef __attribute__((address_space(1))) v4i as1_v4i;   // global v4i32
typedef __attribute__((address_space(3))) v4i as3_v4i;   // LDS v4i32

__global__ __launch_bounds__(THREADS)
void dequant_q4_0_kernel(const uint8_t* __restrict__ raw, float* __restrict__ out)
{
    __shared__ uint32_t smem[CHUNK_DWORDS];

    const uint32_t tid = threadIdx.x;
    const uint64_t wg  = blockIdx.x;
    const uint8_t* src = raw + wg * (uint64_t)CHUNK_BYTES;   // 16B-aligned (4608 = 288*16)

    // ---- Stage 4608 B of raw blocks into LDS via CDNA5 async global->LDS DMA ----
#if __has_builtin(__builtin_amdgcn_global_load_async_to_lds_b128)
    {
        as1_v4i* g0 = (as1_v4i*)(src + tid * 16);
        as3_v4i* l0 = (as3_v4i*)((char*)smem + tid * 16);
        __builtin_amdgcn_global_load_async_to_lds_b128(g0, l0, 0, 0);
        if (tid < (CHUNK_BYTES - THREADS * 16) / 16) {       // wave 0 only: last 512 B
            as1_v4i* g1 = (as1_v4i*)(src + THREADS * 16 + tid * 16);
            as3_v4i* l1 = (as3_v4i*)((char*)smem + THREADS * 16 + tid * 16);
            __builtin_amdgcn_global_load_async_to_lds_b128(g1, l1, 0, 0);
        }
    }
#if __has_builtin(__builtin_amdgcn_s_wait_asynccnt)
    __builtin_amdgcn_s_wait_asynccnt(0);
#else
    asm volatile("s_wait_asynccnt 0" ::: "memory");
#endif
#else
    {
        const u32x4* s4 = (const u32x4*)src;
        u32x4*       d4 = (u32x4*)smem;
        d4[tid] = s4[tid];
        if (tid < (CHUNK_BYTES - THREADS * 16) / 16)
            d4[THREADS + tid] = s4[THREADS + tid];
    }
#endif
    __syncthreads();

    // ---- Decode one block per thread ----
    const uint32_t base = tid * BLOCK_BYTES;      // block start byte, ==0 or 2 (mod 4)
    const uint32_t a    = base >> 2;              // dword index of aligned 20B window
    const uint32_t D0 = smem[a + 0];
    const uint32_t D1 = smem[a + 1];
    const uint32_t D2 = smem[a + 2];
    const uint32_t D3 = smem[a + 3];
    const uint32_t D4 = smem[a + 4];

    const uint32_t half  = (base & 2u) << 3;      // 0 or 16
    const uint16_t sbits = (uint16_t)(D0 >> half);
    const float d   = (float)__builtin_bit_cast(_Float16, sbits);
    const float m8d = -8.0f * d;

    // Extract quant bytes (block bytes 2..17) with a 64-bit funnel shift.
    const uint32_t sh = 16u + half;               // 16 or 32
    uint32_t q[4];
    q[0] = (uint32_t)((((uint64_t)D1 << 32) | D0) >> sh);
    q[1] = (uint32_t)((((uint64_t)D2 << 32) | D1) >> sh);
    q[2] = (uint32_t)((((uint64_t)D3 << 32) | D2) >> sh);
    q[3] = (uint32_t)((((uint64_t)D4 << 32) | D3) >> sh);

    f32x4* dst4 = (f32x4*)(out + (wg * (uint64_t)BLOCKS_PER_WG + tid) * QK);

#pragma unroll
    for (int k = 0; k < 4; ++k) {
        const uint32_t lo = q[k] & 0x0F0F0F0Fu;          // out[4k .. 4k+3]
        const uint32_t hi = (q[k] >> 4) & 0x0F0F0F0Fu;   // out[16+4k .. 16+4k+3]
        f32x4 vlo, vhi;
        vlo.x = fmaf(d, (float)( lo        & 0xFFu), m8d);
        vlo.y = fmaf(d, (float)((lo >>  8) & 0xFFu), m8d);
        vlo.z = fmaf(d, (float)((lo >> 16) & 0xFFu), m8d);
        vlo.w = fmaf(d, (float)( lo >> 24        ), m8d);
        vhi.x = fmaf(d, (float)( hi        & 0xFFu), m8d);
        vhi.y = fmaf(d, (float)((hi >>  8) & 0xFFu), m8d);
        vhi.z = fmaf(d, (float)((hi >> 16) & 0xFFu), m8d);
        vhi.w = fmaf(d, (float)( hi >> 24        ), m8d);
        __builtin_nontemporal_store(vlo, dst4 + k);      // stream 256 MB past L2
        __builtin_nontemporal_store(vhi, dst4 + 4 + k);
    }
}

extern "C" void kernel_launch(void* const* d_in, const int* in_sizes, int n_in,
                              void* d_out, int out_size, void* d_ws, size_t ws_size,
                              hipStream_t stream)
{
    const uint8_t* raw = (const uint8_t*)d_in[0];
    float*         out = (float*)d_out;

    const int nblocks = in_sizes[0] / BLOCK_BYTES;          // 2,097,152
    const int nwg     = nblocks / BLOCKS_PER_WG;            // 8192 (exact)

    dequant_q4_0_kernel<<<nwg, THREADS, 0, stream>>>(raw, out);
}